// MaskLoss_34643206209975
// MI455X (gfx1250) — compile-verified
//
#include <hip/hip_runtime.h>
#include <stdint.h>

// Fixed reference shapes
#define B_     2
#define C_     32
#define A_     2100
#define APAD   2112          // A padded to multiple of 16
#define MT     132           // APAD/16
#define NMASK  32
#define HP     80
#define HW     6400          // 80*80
#define NTPIX  400           // HW/16 (5 tiles per pixel row)
#define DIVT   (MT*MT)       // 17424 tiles per batch
#define HGT    320
#define WGT    320

typedef float v2f __attribute__((ext_vector_type(2)));
typedef float v8f __attribute__((ext_vector_type(8)));

// D = A(16x4 f32) * B(4x16 f32) + C(16x16 f32) — full f32 precision matrix core
__device__ __forceinline__ v8f wmma_k4(v2f a, v2f b, v8f c) {
  return __builtin_amdgcn_wmma_f32_16x16x4_f32(false, a, false, b, (short)0, c,
                                               false, false);
}

// ---------------------------------------------------------------------------
// 1) coeffs[b, a, c] = concat(p0,p1,p2) flattened B C H W -> B (HW) C,
//    padded with zeros to APAD rows; cn = coeffs / max(||coeffs||, 1e-12)
// ---------------------------------------------------------------------------
__global__ void build_coeffs_kernel(const float* __restrict__ p0,
                                    const float* __restrict__ p1,
                                    const float* __restrict__ p2,
                                    float* __restrict__ coeffs,
                                    float* __restrict__ cn) {
  int lane = threadIdx.x & 31;      // channel c
  int la   = threadIdx.x >> 5;      // local anchor (8 per block)
  int a    = blockIdx.x * 8 + la;
  int b    = blockIdx.y;
  if (a >= APAD) return;
  float v = 0.f;
  if (a < A_) {
    if (a < 1600)      v = p0[((long)b * C_ + lane) * 1600 + a];
    else if (a < 2000) v = p1[((long)b * C_ + lane) * 400 + (a - 1600)];
    else               v = p2[((long)b * C_ + lane) * 100 + (a - 2000)];
  }
  long off = ((long)b * APAD + a) * C_ + lane;
  coeffs[off] = v;
  float s = v * v;
  for (int m = 1; m < 32; m <<= 1) s += __shfl_xor(s, m, 32);
  cn[off] = v / fmaxf(sqrtf(s), 1e-12f);
}

// ---------------------------------------------------------------------------
// 2) 320->80 bilinear (align_corners=False; frac=0.5 -> 2x2 average),
//    binarize > 0.5, and per-mask pixel area.
// ---------------------------------------------------------------------------
__global__ void downsample_gt_kernel(const float* __restrict__ gt,
                                     float* __restrict__ down,
                                     float* __restrict__ areas) {
  int bn = blockIdx.x;  // b*NMASK + n
  const float* src = gt + (long)bn * HGT * WGT;
  float* dst = down + (long)bn * HW;
  float cnt = 0.f;
  for (int p = threadIdx.x; p < HW; p += blockDim.x) {
    int y = p / HP, x = p % HP;
    const float* r0 = src + (4 * y + 1) * WGT + (4 * x + 1);
    float avg = 0.25f * (r0[0] + r0[1] + r0[WGT] + r0[WGT + 1]);
    float v = avg > 0.5f ? 1.f : 0.f;
    dst[p] = v;
    cnt += v;
  }
  __shared__ float red[256];
  red[threadIdx.x] = cnt;
  __syncthreads();
  for (int s = 128; s > 0; s >>= 1) {
    if ((int)threadIdx.x < s) red[threadIdx.x] += red[threadIdx.x + s];
    __syncthreads();
  }
  if (threadIdx.x == 0) areas[bn] = red[0];
}

// ---------------------------------------------------------------------------
// 3) Fused GEMM + BCE epilogue: per_mask[b,a] = sum over pixels-in-box of
//    BCE(coeffs[a]·protos[:,p], down_gt[idx[a],p]).
//    Block = one 16-anchor tile, 8 waves sweep the 400 pixel tiles.
//    Epilogue is fully branchless (unconditional clamped gathers, float mask).
// ---------------------------------------------------------------------------
__global__ void mask_loss_gemm_kernel(const float* __restrict__ coeffs,
                                      const float* __restrict__ protos,
                                      const float* __restrict__ down_gt,
                                      const long long* __restrict__ mask_idx,
                                      const float* __restrict__ boxes,
                                      float* __restrict__ per_mask) {
  int mt = blockIdx.x;
  int b  = blockIdx.y;
  int lane = threadIdx.x & 31;
  int wv   = threadIdx.x >> 5;   // 0..7
  int m_base = mt * 16;

  __shared__ int   s_goff[16];      // idx * HW (precomputed gather base)
  __shared__ float s_box[16][4];
  __shared__ float s_part[8][16];

  if (threadIdx.x < 16) {
    int a = m_base + (int)threadIdx.x;
    int idx = 0;
    // padded/out-of-range anchors get an empty box -> mask contribution 0
    float x1 = 1e9f, y1 = 1e9f, x2 = -1e9f, y2 = -1e9f;
    if (a < A_) {
      idx = (int)mask_idx[(long)b * A_ + a];
      const float* bp = boxes + ((long)b * A_ + a) * 4;
      x1 = bp[0] * 0.25f; y1 = bp[1] * 0.25f;   // proto-coord scale 80/320
      x2 = bp[2] * 0.25f; y2 = bp[3] * 0.25f;
    }
    s_goff[threadIdx.x] = idx * HW;
    s_box[threadIdx.x][0] = x1; s_box[threadIdx.x][1] = y1;
    s_box[threadIdx.x][2] = x2; s_box[threadIdx.x][3] = y2;
  }
  __syncthreads();

  const float* cA = coeffs + (long)b * APAD * C_;
  const float* P  = protos + (long)b * C_ * HW;
  const float* G  = down_gt + (long)b * NMASK * HW;

  int rl = lane & 15, hi = lane >> 4;

  // A fragments: coeffs tile 16x32, hoisted (8 x v2f per lane)
  v2f afr[8];
  {
    const float* ap = cA + (long)(m_base + rl) * C_ + 2 * hi;
#pragma unroll
    for (int k = 0; k < 8; ++k) { afr[k].x = ap[4 * k]; afr[k].y = ap[4 * k + 1]; }
  }

  float rs[8];
#pragma unroll
  for (int r = 0; r < 8; ++r) rs[r] = 0.f;

  for (int nt = wv; nt < NTPIX; nt += 8) {
    // nt is wave-uniform: 5 tiles per 80-pixel row -> scalar x/y math
    int yrow  = nt / 5;
    int xbase = (nt - yrow * 5) * 16;
    int col   = nt * 16 + rl;
    float xf = (float)(xbase + rl);
    float yf = (float)yrow;

    v2f bfr[8];
#pragma unroll
    for (int k = 0; k < 8; ++k) {
      int kr = 4 * k + 2 * hi;
      bfr[k].x = P[(long)kr * HW + col];
      bfr[k].y = P[(long)(kr + 1) * HW + col];
    }
    v8f acc = {};
#pragma unroll
    for (int k = 0; k < 8; ++k) acc = wmma_k4(afr[k], bfr[k], acc);

#pragma unroll
    for (int r = 0; r < 8; ++r) {
      int rr = r + 8 * hi;
      float z  = acc[r];
      float gv = G[s_goff[rr] + col];                  // unconditional gather
      float inb = ((xf >= s_box[rr][0]) & (xf < s_box[rr][2]) &
                   (yf >= s_box[rr][1]) & (yf < s_box[rr][3])) ? 1.f : 0.f;
      float bce = fmaxf(z, 0.f) - z * gv + log1pf(expf(-fabsf(z)));
      rs[r] = fmaf(inb, bce, rs[r]);
    }
  }

  // reduce across the 16 lanes of each half-wave (same rows share a half)
#pragma unroll
  for (int r = 0; r < 8; ++r) {
    float v = rs[r];
    for (int m = 1; m < 16; m <<= 1) v += __shfl_xor(v, m, 32);
    rs[r] = v;
  }
  if (rl == 0) {
#pragma unroll
    for (int r = 0; r < 8; ++r) s_part[wv][r + 8 * hi] = rs[r];
  }
  __syncthreads();
  if (threadIdx.x < 16) {
    float s = 0.f;
#pragma unroll
    for (int w = 0; w < 8; ++w) s += s_part[w][threadIdx.x];
    per_mask[(long)b * APAD + m_base + threadIdx.x] = s;
  }
}

// ---------------------------------------------------------------------------
// 4) Diversity: one wave per 16x16 anchor-pair tile of cos_sim = cn · cnT,
//    fused loss epilogue; i-side metadata fetched once per lane and
//    distributed by wave shuffles (no per-row gathers).
// ---------------------------------------------------------------------------
__global__ void diversity_gemm_kernel(const float* __restrict__ cn,
                                      const long long* __restrict__ mask_idx,
                                      const unsigned char* __restrict__ valid,
                                      float* __restrict__ div_part) {
  int jt = blockIdx.x, it = blockIdx.y, b = blockIdx.z;
  int lane = threadIdx.x;           // blockDim == 32 (one wave)
  int rl = lane & 15, hi = lane >> 4;
  const float* cb = cn + (long)b * APAD * C_;

  v2f afr[8], bfr[8];
  {
    const float* ap = cb + (long)(it * 16 + rl) * C_ + 2 * hi;
    const float* bp = cb + (long)(jt * 16 + rl) * C_ + 2 * hi;  // cnT loads same pattern
#pragma unroll
    for (int k = 0; k < 8; ++k) {
      afr[k].x = ap[4 * k]; afr[k].y = ap[4 * k + 1];
      bfr[k].x = bp[4 * k]; bfr[k].y = bp[4 * k + 1];
    }
  }
  v8f acc = {};
#pragma unroll
  for (int k = 0; k < 8; ++k) acc = wmma_k4(afr[k], bfr[k], acc);

  // j-side (column) metadata: one per lane
  int j = jt * 16 + rl;
  bool jok = j < A_;
  int   ij = jok ? (int)mask_idx[(long)b * A_ + j] : -1;
  float vj = (jok && valid[(long)b * A_ + j]) ? 1.f : 0.f;

  // i-side (row) metadata: lane rl holds values for anchor it*16+rl
  int i_l = it * 16 + rl;
  bool iok = i_l < A_;
  int   ii_l = iok ? (int)mask_idx[(long)b * A_ + i_l] : -2;
  float vi_l = (iok && valid[(long)b * A_ + i_l]) ? 1.f : 0.f;

  float s = 0.f;
#pragma unroll
  for (int r = 0; r < 8; ++r) {
    int rr = r + 8 * hi;                 // row index within the 16-row tile
    int   ii = __shfl(ii_l, rr, 32);
    float vi = __shfl(vi_l, rr, 32);
    float cs = (acc[r] + 1.f) * 0.5f;
    float lm = (ii == ij) ? (1.f - cs) : cs;
    s += lm * vi * vj;
  }
  for (int m = 1; m < 32; m <<= 1) s += __shfl_xor(s, m, 32);
  if (lane == 0) div_part[((long)b * MT + it) * MT + jt] = s;
}

// ---------------------------------------------------------------------------
// 5) Finalize both scalars in one block (deterministic, no atomics).
// ---------------------------------------------------------------------------
__global__ void finalize_kernel(const float* __restrict__ per_mask,
                                const float* __restrict__ areas,
                                const long long* __restrict__ mask_idx,
                                const unsigned char* __restrict__ valid,
                                const float* __restrict__ box_norm,
                                const float* __restrict__ cls_norm,
                                const float* __restrict__ div_part,
                                float* __restrict__ out) {
  __shared__ float red[256];
  int t = threadIdx.x;

  float s = 0.f;
  for (int i = t; i < B_ * A_; i += 256) {
    int b = i / A_, a = i % A_;
    if (valid[i]) {
      int idx = (int)mask_idx[i];
      float area = areas[b * NMASK + idx];
      s += per_mask[(long)b * APAD + a] / (area + 1.f) * box_norm[i];
    }
  }
  red[t] = s; __syncthreads();
  for (int k = 128; k > 0; k >>= 1) { if (t < k) red[t] += red[t + k]; __syncthreads(); }
  float mask_loss = red[0] / cls_norm[0];
  __syncthreads();

  float dtot = 0.f;
  for (int b = 0; b < B_; ++b) {
    float nv = 0.f;
    for (int i = t; i < A_; i += 256) nv += valid[(long)b * A_ + i] ? 1.f : 0.f;
    red[t] = nv; __syncthreads();
    for (int k = 128; k > 0; k >>= 1) { if (t < k) red[t] += red[t + k]; __syncthreads(); }
    float nvb = red[0]; __syncthreads();

    float ds = 0.f;
    for (int i = t; i < DIVT; i += 256) ds += div_part[(long)b * DIVT + i];
    red[t] = ds; __syncthreads();
    for (int k = 128; k > 0; k >>= 1) { if (t < k) red[t] += red[t + k]; __syncthreads(); }
    float dsb = red[0]; __syncthreads();

    if (nvb > 0.f) { float d = fmaxf(nvb, 1.f); dtot += dsb / (d * d); }
  }
  if (t == 0) { out[0] = mask_loss; out[1] = dtot / (float)B_; }
}

// ---------------------------------------------------------------------------
extern "C" void kernel_launch(void* const* d_in, const int* in_sizes, int n_in,
                              void* d_out, int out_size, void* d_ws, size_t ws_size,
                              hipStream_t stream) {
  (void)in_sizes; (void)n_in; (void)out_size; (void)ws_size;
  const float*         p0       = (const float*)d_in[0];
  const float*         p1       = (const float*)d_in[1];
  const float*         p2       = (const float*)d_in[2];
  const float*         protos   = (const float*)d_in[3];
  const float*         gt       = (const float*)d_in[4];
  const long long*     mask_idx = (const long long*)d_in[5];    // int64
  const unsigned char* valid    = (const unsigned char*)d_in[6]; // bool
  const float*         boxes    = (const float*)d_in[7];
  const float*         box_norm = (const float*)d_in[8];
  const float*         cls_norm = (const float*)d_in[9];
  float* out = (float*)d_out;

  // workspace layout (floats)
  float* ws = (float*)d_ws;
  float* coeffs   = ws;                               // B*APAD*C = 135168
  float* cn       = coeffs + (long)B_ * APAD * C_;    // 135168
  float* down_gt  = cn + (long)B_ * APAD * C_;        // B*NMASK*HW = 409600
  float* areas    = down_gt + (long)B_ * NMASK * HW;  // B*NMASK = 64
  float* per_mask = areas + (long)B_ * NMASK;         // B*APAD = 4224
  float* div_part = per_mask + (long)B_ * APAD;       // B*DIVT = 34848

  build_coeffs_kernel<<<dim3(APAD / 8, B_), 256, 0, stream>>>(p0, p1, p2, coeffs, cn);
  downsample_gt_kernel<<<B_ * NMASK, 256, 0, stream>>>(gt, down_gt, areas);
  mask_loss_gemm_kernel<<<dim3(MT, B_), 256, 0, stream>>>(coeffs, protos, down_gt,
                                                          mask_idx, boxes, per_mask);
  diversity_gemm_kernel<<<dim3(MT, MT, B_), 32, 0, stream>>>(cn, mask_idx, valid,
                                                             div_part);
  finalize_kernel<<<1, 256, 0, stream>>>(per_mask, areas, mask_idx, valid,
                                         box_norm, cls_norm, div_part, out);
}